// gc_component_32006096290306
// MI455X (gfx1250) — compile-verified
//
#include <hip/hip_runtime.h>
#include <hip/hip_bf16.h>
#include <stddef.h>

// CDNA5 fp32 WMMA operand types: A/B of 16x16x4_f32 are 2 VGPRs, C/D are 8.
typedef __attribute__((ext_vector_type(2))) float v2f;
typedef __attribute__((ext_vector_type(8))) float v8f;
// Tensor DMA descriptor groups (this toolchain: 6-arg clang-23 form:
//   (uint32x4 g0, int32x8 g1, int32x4, int32x4, int32x8, i32 cpol)).
typedef __attribute__((ext_vector_type(4))) unsigned int v4u;
typedef __attribute__((ext_vector_type(8))) int v8i;
typedef __attribute__((ext_vector_type(4))) int v4i;

namespace {
constexpr int Bn = 64, Cn = 3, Tn = 300, Vn = 25;
constexpr int CCn = 100, TSn = 25, TLn = 9;
constexpr int Ln = Tn - TLn + 1;          // 292
constexpr int NS = Bn * Tn * Vn;          // 480000 sites
constexpr int TV = Tn * Vn;               // 7500
constexpr float EPS = 1e-5f;
constexpr int NBX = 480;                  // x-moment partial blocks
constexpr int NBZ = 240;                  // z1-stat partial blocks

// ---- workspace layout (float offsets), with aliasing to cap footprint ----
constexpr size_t OFF_Z1  = 0;                                  // 61,440,000 (z1; reused as q0)
constexpr size_t SZ_Z1   = (size_t)NS * 128;
constexpr size_t OFF_Q0  = OFF_Z1;                             // 46,720,000 used (z1 dead)
constexpr size_t OFF_R0  = OFF_Z1 + SZ_Z1;                     // r0 (30.72M) then p1 (46.72M)
constexpr size_t OFF_P1  = OFF_R0;
constexpr size_t SZ_P1   = (size_t)Bn * TSn * CCn * Ln;        // 46,720,000
constexpr size_t OFF_Z2  = OFF_R0 + SZ_P1;                     // 480,000
constexpr size_t OFF_HT  = OFF_Z2 + NS;                        // 480,000 (x_series, [b][v][t])
constexpr size_t OFF_CM  = OFF_HT + NS;                        // 4,000,000
constexpr size_t SZ_CM   = (size_t)Bn * CCn * Vn * Vn;
constexpr size_t OFF_Q1  = OFF_CM + SZ_CM;                     // 4,672,000
constexpr size_t SZ_Q1   = (size_t)Bn * TSn * 10 * Ln;
constexpr size_t OFF_MH  = OFF_Q1 + SZ_Q1;                     // 1600
constexpr size_t OFF_N1  = OFF_MH + 1600;                      // 40000
constexpr size_t OFF_WSUM= OFF_N1 + 40000;                     // 2500
constexpr size_t OFF_W0F = OFF_WSUM + 2500;                    // 192
constexpr size_t OFF_B0F = OFF_W0F + 192;                      // 64
constexpr size_t OFF_S1  = OFF_B0F + 64;                       // 128
constexpr size_t OFF_B1F = OFF_S1 + 128;                       // 128
constexpr size_t OFF_SB2 = OFF_B1F + 128;                      // 2
constexpr size_t OFF_XP  = OFF_SB2 + 2;                        // NBX*9
constexpr size_t OFF_ZPS = OFF_XP + (size_t)NBX * 9;           // NBZ*128
constexpr size_t OFF_ZPQ = OFF_ZPS + (size_t)NBZ * 128;        // NBZ*128
constexpr size_t OFF_Z2P = OFF_ZPQ + (size_t)NBZ * 128;        // NS (sum,sq pairs)
constexpr size_t OFF_LP  = OFF_Z2P + NS;                       // 1600
} // namespace

// ---------- x channel moments (deterministic two-stage) ----------
__global__ void k_xmom(const float* __restrict__ x, float* __restrict__ xp) {
  __shared__ float sm[256];
  float a[9];
  for (int k = 0; k < 9; ++k) a[k] = 0.f;
  for (int s = blockIdx.x * 256 + threadIdx.x; s < NS; s += NBX * 256) {
    int b = s / TV, r = s % TV;
    const float* xb = x + (size_t)b * (Cn * TV) + r;
    float x0 = xb[0], x1 = xb[TV], x2 = xb[2 * TV];
    a[0] += x0; a[1] += x1; a[2] += x2;
    a[3] += x0 * x0; a[4] += x0 * x1; a[5] += x0 * x2;
    a[6] += x1 * x1; a[7] += x1 * x2; a[8] += x2 * x2;
  }
  for (int k = 0; k < 9; ++k) {
    sm[threadIdx.x] = a[k]; __syncthreads();
    for (int off = 128; off > 0; off >>= 1) {
      if (threadIdx.x < off) sm[threadIdx.x] += sm[threadIdx.x + off];
      __syncthreads();
    }
    if (threadIdx.x == 0) xp[blockIdx.x * 9 + k] = sm[0];
    __syncthreads();
  }
}

// ---------- analytic BN0 fold: (fb0_w,fb0_b,bn0) -> folded affine ----------
__global__ void k_bn0fold(const float* __restrict__ xp, const float* __restrict__ w0,
                          const float* __restrict__ b0, const float* __restrict__ g0,
                          const float* __restrict__ bt0, float* __restrict__ W0f,
                          float* __restrict__ b0f) {
  __shared__ float mom[9];
  if (threadIdx.x < 9) {
    float s = 0.f;
    for (int b = 0; b < NBX; ++b) s += xp[b * 9 + threadIdx.x];
    mom[threadIdx.x] = s;
  }
  __syncthreads();
  int o = threadIdx.x;  // 64 threads
  float inv = 1.f / (float)NS;
  float m0 = mom[0]*inv, m1 = mom[1]*inv, m2 = mom[2]*inv;
  float M00 = mom[3]*inv, M01 = mom[4]*inv, M02 = mom[5]*inv;
  float M11 = mom[6]*inv, M12 = mom[7]*inv, M22 = mom[8]*inv;
  float wa = w0[o*3], wb = w0[o*3+1], wc = w0[o*3+2], bb = b0[o];
  float lin = wa*m0 + wb*m1 + wc*m2;
  float mu  = lin + bb;
  float e2  = wa*wa*M00 + wb*wb*M11 + wc*wc*M22
            + 2.f*(wa*wb*M01 + wa*wc*M02 + wb*wc*M12)
            + 2.f*bb*lin + bb*bb;
  float var = e2 - mu*mu;
  float s   = g0[o] * rsqrtf(var + EPS);
  W0f[o*3] = s*wa; W0f[o*3+1] = s*wb; W0f[o*3+2] = s*wc;
  b0f[o]   = s*bb + bt0[o] - s*mu;
}

// ---------- layer0: r0[site][64] = relu(folded affine of x) ----------
__global__ void k_fb0(const float* __restrict__ x, const float* __restrict__ W0f,
                      const float* __restrict__ b0f, float* __restrict__ r0) {
  __shared__ float xs[4][3];
  int tid = threadIdx.x;
  int s0 = blockIdx.x * 4;
  if (tid < 12) {
    int si = tid / 3, c = tid % 3;
    int s = s0 + si, b = s / TV, r = s % TV;
    xs[si][c] = x[(size_t)b * (Cn * TV) + c * TV + r];
  }
  __syncthreads();
  int si = tid >> 6, o = tid & 63;
  float v = b0f[o] + W0f[o*3]*xs[si][0] + W0f[o*3+1]*xs[si][1] + W0f[o*3+2]*xs[si][2];
  r0[(size_t)(s0 + si) * 64 + o] = fmaxf(v, 0.f);
}

// ---------- fb1: z1 = r0 * W1^T via v_wmma_f32_16x16x4_f32 (bias folded later) ----------
__global__ void __launch_bounds__(256) k_fb1(const float* __restrict__ r0,
                                             const float* __restrict__ w1,
                                             float* __restrict__ z1) {
  __shared__ float lw[64 * 128];  // W1^T in LDS, [k][n]
  int tid = threadIdx.x;
  for (int idx = tid; idx < 8192; idx += 256) {
    int o = idx & 127, c = idx >> 7;
    lw[c * 128 + o] = w1[o * 64 + c];
  }
  __syncthreads();
  int wave = tid >> 5, lane = tid & 31, hl = lane >> 4, ln = lane & 15;
  int nb = wave * 16;               // 8 waves = 8 n-tiles of 128
  int m0 = blockIdx.x * 16;
  const float* arow = r0 + (size_t)(m0 + ln) * 64;
  v8f acc = {0.f,0.f,0.f,0.f,0.f,0.f,0.f,0.f};
  for (int k0 = 0; k0 < 64; k0 += 4) {
    int k = k0 + 2 * hl;
    v2f a = *(const v2f*)(arow + k);
    v2f b; b.x = lw[k * 128 + nb + ln]; b.y = lw[(k + 1) * 128 + nb + ln];
    acc = __builtin_amdgcn_wmma_f32_16x16x4_f32(false, a, false, b, (short)0, acc, false, false);
  }
  int col = nb + ln;
  for (int r = 0; r < 8; ++r)
    z1[(size_t)(m0 + r + 8 * hl) * 128 + col] = acc[r];
}

// ---------- per-channel z1 sum/sumsq partials ----------
__global__ void k_z1p(const float* __restrict__ z1, float* __restrict__ zps,
                      float* __restrict__ zpq) {
  __shared__ float sm[256];
  int tid = threadIdx.x, o = tid & 127, j = tid >> 7;
  float S = 0.f, Q = 0.f;
  for (int s = blockIdx.x * 2 + j; s < NS; s += NBZ * 2) {
    float v = z1[(size_t)s * 128 + o];
    S += v; Q += v * v;
  }
  sm[tid] = S; __syncthreads();
  if (j == 0) zps[blockIdx.x * 128 + o] = sm[o] + sm[o + 128];
  __syncthreads();
  sm[tid] = Q; __syncthreads();
  if (j == 0) zpq[blockIdx.x * 128 + o] = sm[o] + sm[o + 128];
}

__global__ void k_bn1fold(const float* __restrict__ zps, const float* __restrict__ zpq,
                          const float* __restrict__ b1, const float* __restrict__ g1,
                          const float* __restrict__ bt1, float* __restrict__ s1,
                          float* __restrict__ b1f) {
  int o = threadIdx.x;  // 128
  float S = 0.f, Q = 0.f;
  for (int b = 0; b < NBZ; ++b) { S += zps[b * 128 + o]; Q += zpq[b * 128 + o]; }
  float bb = b1[o];
  float mu = S / NS + bb;                       // stored z1 excludes bias
  float e2 = Q / NS + 2.f * bb * S / NS + bb * bb;
  float var = e2 - mu * mu;
  float s = g1[o] * rsqrtf(var + EPS);
  s1[o] = s;
  b1f[o] = s * bb + bt1[o] - s * mu;
}

// ---------- r1=relu(bn1(z1)) dotted with fb2_w -> z2 (+ stats) ----------
__global__ void k_z2(const float* __restrict__ z1, const float* __restrict__ s1,
                     const float* __restrict__ b1f, const float* __restrict__ w2,
                     float* __restrict__ z2s, float* __restrict__ z2p) {
  __shared__ float sm[256];
  int tid = threadIdx.x, o = tid & 127, j = tid >> 7;
  int s = blockIdx.x * 2 + j;
  float r = fmaxf(s1[o] * z1[(size_t)s * 128 + o] + b1f[o], 0.f);
  sm[tid] = w2[o] * r;
  __syncthreads();
  for (int off = 64; off > 0; off >>= 1) {
    if (o < off) sm[tid] += sm[tid + off];
    __syncthreads();
  }
  if (tid == 0) {
    float a = sm[0], b = sm[128];
    z2s[blockIdx.x * 2] = a; z2s[blockIdx.x * 2 + 1] = b;
    z2p[blockIdx.x * 2] = a + b; z2p[blockIdx.x * 2 + 1] = a * a + b * b;
  }
}

__global__ void k_bn2fold(const float* __restrict__ z2p, const float* __restrict__ b2p,
                          const float* __restrict__ g2p, const float* __restrict__ bt2p,
                          float* __restrict__ sb2) {
  __shared__ float sm[256], sq[256];
  float S = 0.f, Q = 0.f;
  for (int i = threadIdx.x; i < NS / 2; i += 256) { S += z2p[i * 2]; Q += z2p[i * 2 + 1]; }
  sm[threadIdx.x] = S; sq[threadIdx.x] = Q; __syncthreads();
  for (int off = 128; off > 0; off >>= 1) {
    if (threadIdx.x < off) { sm[threadIdx.x] += sm[threadIdx.x + off]; sq[threadIdx.x] += sq[threadIdx.x + off]; }
    __syncthreads();
  }
  if (threadIdx.x == 0) {
    float b2 = b2p[0];
    float mu = sm[0] / NS + b2;
    float e2 = sq[0] / NS + 2.f * b2 * sm[0] / NS + b2 * b2;
    float var = e2 - mu * mu;
    float s = g2p[0] * rsqrtf(var + EPS);
    sb2[0] = s; sb2[1] = s * b2 + bt2p[0] - s * mu;
  }
}

// ---------- h -> x_series transposed layout ht[b][v][t] ----------
__global__ void k_ht(const float* __restrict__ z2s, const float* __restrict__ sb2,
                     float* __restrict__ ht) {
  int s = blockIdx.x * 256 + threadIdx.x;
  float v = fmaxf(sb2[0] * z2s[s] + sb2[1], 0.f);
  int b = s / TV, r = s % TV, t = r / Vn, vv = r % Vn;
  ht[(size_t)b * TV + vv * Tn + t] = v;
}

__global__ void k_mh(const float* __restrict__ ht, float* __restrict__ mh) {
  int i = blockIdx.x * 256 + threadIdx.x;
  if (i >= Bn * Vn) return;
  const float* p = ht + (size_t)i * Tn;
  float s = 0.f;
  for (int t = 0; t < Tn; ++t) s += p[t];
  mh[i] = s / (float)Tn;
}

// ---------- cm = tanh(x1[b,c,i] - x2[b,c,j]) ----------
__global__ void k_cm(const float* __restrict__ mh, const float* __restrict__ c1w,
                     const float* __restrict__ c1b, const float* __restrict__ c2w,
                     const float* __restrict__ c2b, float* __restrict__ cm) {
  int idx = blockIdx.x * 256 + threadIdx.x;      // exactly 4,000,000
  int b = idx / 62500, rem = idx % 62500;
  int c = rem / 625, ij = rem % 625;
  int i = ij / 25, j = ij % 25;
  float v = c1w[c] * mh[b * 25 + i] + c1b[c] - c2w[c] * mh[b * 25 + j] - c2b[c];
  cm[idx] = tanhf(v);
}

__global__ void k_wsum(const float* __restrict__ w, float* __restrict__ wsum) {
  int i = blockIdx.x * 256 + threadIdx.x;
  if (i >= CCn * TSn) return;
  const float* p = w + (size_t)i * TLn;
  float s = 0.f;
  for (int l = 0; l < TLn; ++l) s += p[l] * p[l];
  wsum[i] = s;
}

// ---------- GCs + n1 (both reduce over CC) ----------
__global__ void k_gcs(const float* __restrict__ cm, const float* __restrict__ wsum,
                      float* __restrict__ gcs, float* __restrict__ n1) {
  int idx = blockIdx.x * 256 + threadIdx.x;
  if (idx >= Bn * Vn * Vn) return;
  int b = idx / 625, vm = idx % 625, m = vm % 25;
  const float* p = cm + (size_t)b * 62500 + vm;
  float sA = 0.f, sB = 0.f;
  for (int c = 0; c < CCn; ++c) {
    float v = p[c * 625], v2 = v * v;
    sB += v2; sA += v2 * wsum[c * 25 + m];
  }
  gcs[idx] = sqrtf(sA);
  n1[idx] = sqrtf(sB);
}

__global__ void k_pan(const float* __restrict__ n1, float* __restrict__ out) {
  __shared__ float s1m[256], s2m[256];
  float a1 = 0.f, a2 = 0.f;
  for (int p = threadIdx.x; p < Bn * Vn; p += 256) {
    int b = p / 25, j = p % 25;
    const float* q = n1 + (size_t)b * 625 + j;
    float cs = 0.f, ss = 0.f;
    for (int i = 0; i < 25; ++i) { float v = q[i * 25]; cs += v * v; ss += v; }
    a1 += sqrtf(cs); a2 += ss;
  }
  s1m[threadIdx.x] = a1; s2m[threadIdx.x] = a2; __syncthreads();
  for (int off = 128; off > 0; off >>= 1) {
    if (threadIdx.x < off) { s1m[threadIdx.x] += s1m[threadIdx.x + off]; s2m[threadIdx.x] += s2m[threadIdx.x + off]; }
    __syncthreads();
  }
  if (threadIdx.x == 0) out[0] = 1e-4f * (s1m[0] + s2m[0]);
}

__global__ void k_reg(const float* __restrict__ f0w, const float* __restrict__ f1w,
                      const float* __restrict__ f2w, float* __restrict__ out) {
  __shared__ float sm[256];
  float s = 0.f;
  for (int i = threadIdx.x; i < TSn * CCn * CCn; i += 256) { float v = f0w[i]; s += v * v; }
  for (int i = threadIdx.x; i < TSn * 10 * CCn;  i += 256) { float v = f1w[i]; s += v * v; }
  for (int i = threadIdx.x; i < TSn * 10;        i += 256) { float v = f2w[i]; s += v * v; }
  sm[threadIdx.x] = s; __syncthreads();
  for (int off = 128; off > 0; off >>= 1) {
    if (threadIdx.x < off) sm[threadIdx.x] += sm[threadIdx.x + off];
    __syncthreads();
  }
  if (threadIdx.x == 0) out[0] = 1e-4f * sm[0];
}

// ---------- fused Fc build (in LDS) + p-einsum WMMA: per (b,c) block ----------
__global__ void __launch_bounds__(256) k_fcp(const float* __restrict__ ht,
                                             const float* __restrict__ wgt,
                                             const float* __restrict__ wnorm,
                                             const float* __restrict__ cm,
                                             float* __restrict__ p1) {
  __shared__ float xsL[25 * 304];   // time series rows, zero-padded tail
  __shared__ float FcL[28 * 304];   // K-padded (v=25..27 zero), N-padded (t>=292 zero)
  __shared__ float cmT[32 * 28];    // A[i][v] = cm[b,c,v,i], zero-padded
  __shared__ float wnL[232];
  int tid = threadIdx.x;
  int b = blockIdx.x / CCn, c = blockIdx.x % CCn;
  for (int idx = tid; idx < 25 * 304; idx += 256) {
    int v = idx / 304, t = idx % 304;
    xsL[idx] = (t < Tn) ? ht[(size_t)b * TV + v * Tn + t] : 0.f;
  }
  for (int idx = tid; idx < 225; idx += 256)
    wnL[idx] = wgt[(size_t)c * 225 + idx] / wnorm[idx];
  for (int idx = tid; idx < 32 * 28; idx += 256) cmT[idx] = 0.f;
  __syncthreads();
  for (int idx = tid; idx < 625; idx += 256) {
    int v = idx / 25, i = idx % 25;
    cmT[i * 28 + v] = cm[(size_t)b * 62500 + c * 625 + idx];
  }
  for (int idx = tid; idx < 28 * 304; idx += 256) {
    int v = idx / 304, t = idx % 304;
    float s = 0.f;
    if (v < 25 && t < Ln) {
      for (int l = 0; l < TLn; ++l) s += xsL[v * 304 + t + l] * wnL[v * 9 + l];
    }
    FcL[idx] = s;
  }
  __syncthreads();
  int wave = tid >> 5, lane = tid & 31, hl = lane >> 4, ln = lane & 15;
  for (int tile = wave; tile < 2 * 19; tile += 8) {   // 2 i-tiles x 19 t-tiles
    int i0 = (tile / 19) * 16, t0 = (tile % 19) * 16;
    v8f acc = {0.f,0.f,0.f,0.f,0.f,0.f,0.f,0.f};
    for (int v0 = 0; v0 < 28; v0 += 4) {
      int av = v0 + 2 * hl;
      v2f a = *(const v2f*)&cmT[(i0 + ln) * 28 + av];
      v2f bb; bb.x = FcL[av * 304 + t0 + ln]; bb.y = FcL[(av + 1) * 304 + t0 + ln];
      acc = __builtin_amdgcn_wmma_f32_16x16x4_f32(false, a, false, bb, (short)0, acc, false, false);
    }
    int t = t0 + ln;
    if (t < Ln) {
      for (int r = 0; r < 8; ++r) {
        int i = i0 + r + 8 * hl;
        if (i < TSn)
          p1[(((size_t)b * TSn + i) * CCn + c) * Ln + t] = acc[r];
      }
    }
  }
}

// ---------- f0: per (b,g), 100x100 * 100x292 WMMA, relu+bias ----------
// Weight panel (40 KB) staged LDS-side by the Tensor Data Mover (TENSORcnt),
// then repacked into a zero-padded even-stride WMMA panel.
__global__ void __launch_bounds__(256) k_f0(const float* __restrict__ p1,
                                            const float* __restrict__ f0w,
                                            const float* __restrict__ f0b,
                                            float* __restrict__ q0) {
  __shared__ float stage[CCn * CCn];   // raw 100x100 panel, TDM destination
  __shared__ float wA[112 * 104];      // zero-padded panel, even row stride
  int tid = threadIdx.x;
  int b = blockIdx.x / TSn, g = blockIdx.x % TSn;
  for (int idx = tid; idx < 112 * 104; idx += 256) wA[idx] = 0.f;

  if (tid < 32) {   // one wave issues the DMA (uniform descriptor -> SGPRs)
    unsigned long long ga =
        (unsigned long long)(const void*)(f0w + (size_t)g * CCn * CCn);
    unsigned int lds_off = (unsigned int)(size_t)(void*)stage;  // flat[31:0] = LDS offset
    v4u g0;
    g0[0] = 1u;                                     // count=1, user descriptor
    g0[1] = lds_off;                                // lds_addr
    g0[2] = (unsigned int)(ga & 0xFFFFFFFFu);       // global_addr[31:0]
    g0[3] = (unsigned int)((ga >> 32) & 0x1FFFFFFu) // global_addr[56:32]
          | (2u << 30);                             // type=2 ("image")
    v8i g1;
    g1[0] = 0x20000;                                // data_size=4B; no multicast/pad
    g1[1] = (int)((10000u & 0xFFFFu) << 16);        // tensor_dim0 lo16
    g1[2] = (int)((10000u >> 16) | (1u << 16));     // tensor_dim0 hi16 | tensor_dim1 lo16=1
    g1[3] = (int)(10000u << 16);                    // tensor_dim1 hi16=0 | tile_dim0=10000
    g1[4] = 1;                                      // tile_dim1=1, tile_dim2=0
    g1[5] = 10000;                                  // tensor_dim0_stride lo32
    g1[6] = 0;                                      // stride hi16 | tensor_dim1_stride lo16
    g1[7] = 0;
    v4i g2; g2[0] = 1; g2[1] = 1; g2[2] = 10000; g2[3] = 0;  // dims 2/3 degenerate
    v4i g3; g3[0] = 10000; g3[1] = (1 << 16); g3[2] = 0; g3[3] = 0;
    v8i g4 = {0, 0, 0, 0, 0, 0, 0, 0};             // aux group (unused, flat 1-D tile)
    __builtin_amdgcn_tensor_load_to_lds(g0, g1, g2, g3, g4, 0);
    __builtin_amdgcn_s_wait_tensorcnt(0);           // s_wait_tensorcnt 0
  }
  __syncthreads();
  for (int idx = tid; idx < CCn * CCn; idx += 256) {
    int o = idx / CCn, cc = idx % CCn;
    wA[o * 104 + cc] = stage[idx];
  }
  __syncthreads();
  const float* Pg = p1 + ((size_t)b * TSn + g) * CCn * Ln;
  float* Qg = q0 + ((size_t)b * TSn + g) * CCn * Ln;
  int wave = tid >> 5, lane = tid & 31, hl = lane >> 4, ln = lane & 15;
  for (int tile = wave; tile < 7 * 19; tile += 8) {
    int o0 = (tile / 19) * 16, t0 = (tile % 19) * 16;
    int t = t0 + ln; bool tin = t < Ln;
    int nt = tile + 8;                  // prefetch next tile's B-panel column
    if (nt < 7 * 19) __builtin_prefetch(Pg + (nt % 19) * 16 + ln, 0, 1);
    v8f acc = {0.f,0.f,0.f,0.f,0.f,0.f,0.f,0.f};
    for (int c0 = 0; c0 < CCn; c0 += 4) {
      int ac = c0 + 2 * hl;
      v2f a = *(const v2f*)&wA[(o0 + ln) * 104 + ac];
      v2f bb;
      bb.x = tin ? Pg[(size_t)ac * Ln + t] : 0.f;
      bb.y = tin ? Pg[(size_t)(ac + 1) * Ln + t] : 0.f;
      acc = __builtin_amdgcn_wmma_f32_16x16x4_f32(false, a, false, bb, (short)0, acc, false, false);
    }
    if (tin) {
      for (int r = 0; r < 8; ++r) {
        int o = o0 + r + 8 * hl;
        if (o < CCn) Qg[(size_t)o * Ln + t] = fmaxf(acc[r] + f0b[g * CCn + o], 0.f);
      }
    }
  }
}

// ---------- f1: per (b,g), 10x100 * 100x292 WMMA ----------
__global__ void __launch_bounds__(256) k_f1(const float* __restrict__ q0,
                                            const float* __restrict__ f1w,
                                            const float* __restrict__ f1b,
                                            float* __restrict__ q1) {
  __shared__ float wA[16 * 104];
  int tid = threadIdx.x;
  int b = blockIdx.x / TSn, g = blockIdx.x % TSn;
  for (int idx = tid; idx < 16 * 104; idx += 256) wA[idx] = 0.f;
  __syncthreads();
  for (int idx = tid; idx < 10 * CCn; idx += 256) {
    int o = idx / CCn, cc = idx % CCn;
    wA[o * 104 + cc] = f1w[(size_t)g * 10 * CCn + idx];
  }
  __syncthreads();
  const float* Qg = q0 + ((size_t)b * TSn + g) * CCn * Ln;
  float* Rg = q1 + ((size_t)b * TSn + g) * 10 * Ln;
  int wave = tid >> 5, lane = tid & 31, hl = lane >> 4, ln = lane & 15;
  for (int tile = wave; tile < 19; tile += 8) {
    int t = tile * 16 + ln; bool tin = t < Ln;
    int nt = tile + 8;
    if (nt < 19) __builtin_prefetch(Qg + nt * 16 + ln, 0, 1);
    v8f acc = {0.f,0.f,0.f,0.f,0.f,0.f,0.f,0.f};
    for (int c0 = 0; c0 < CCn; c0 += 4) {
      int ac = c0 + 2 * hl;
      v2f a = *(const v2f*)&wA[ln * 104 + ac];
      v2f bb;
      bb.x = tin ? Qg[(size_t)ac * Ln + t] : 0.f;
      bb.y = tin ? Qg[(size_t)(ac + 1) * Ln + t] : 0.f;
      acc = __builtin_amdgcn_wmma_f32_16x16x4_f32(false, a, false, bb, (short)0, acc, false, false);
    }
    if (tin) {
      for (int r = 0; r < 8; ++r) {
        int o = r + 8 * hl;
        if (o < 10) Rg[(size_t)o * Ln + t] = fmaxf(acc[r] + f1b[g * 10 + o], 0.f);
      }
    }
  }
}

// ---------- f2 + prediction-loss partials (per (b,g)) ----------
__global__ void k_f2loss(const float* __restrict__ q1, const float* __restrict__ f2w,
                         const float* __restrict__ f2b, const float* __restrict__ ht,
                         float* __restrict__ lp) {
  __shared__ float sm[128];
  int tid = threadIdx.x;
  int b = blockIdx.x / TSn, g = blockIdx.x % TSn;
  const float* Rg = q1 + ((size_t)b * TSn + g) * 10 * Ln;
  const float* hrow = ht + (size_t)b * TV + g * Tn;
  float acc = 0.f;
  for (int t = tid; t < Ln - 1; t += 128) {   // t = 291 is discarded by [:-1]
    float s = f2b[g];
    for (int c = 0; c < 10; ++c) s += f2w[g * 10 + c] * Rg[(size_t)c * Ln + t];
    s = fmaxf(s, 0.f);
    float d = hrow[t + TLn] - s;
    acc += d * d;
  }
  sm[tid] = acc; __syncthreads();
  for (int off = 64; off > 0; off >>= 1) {
    if (tid < off) sm[tid] += sm[tid + off];
    __syncthreads();
  }
  if (tid == 0) lp[blockIdx.x] = sm[0];
}

__global__ void k_lossfin(const float* __restrict__ lp, float* __restrict__ out) {
  __shared__ float sm[256];
  float s = 0.f;
  for (int i = threadIdx.x; i < Bn * TSn; i += 256) s += lp[i];
  sm[threadIdx.x] = s; __syncthreads();
  for (int off = 128; off > 0; off >>= 1) {
    if (threadIdx.x < off) sm[threadIdx.x] += sm[threadIdx.x + off];
    __syncthreads();
  }
  if (threadIdx.x == 0) out[0] = sm[0] / (float)(Bn * TSn * (Ln - 1));
}

extern "C" void kernel_launch(void* const* d_in, const int* in_sizes, int n_in,
                              void* d_out, int out_size, void* d_ws, size_t ws_size,
                              hipStream_t stream) {
  (void)in_sizes; (void)n_in; (void)out_size; (void)ws_size;
  const float* x     = (const float*)d_in[0];
  const float* wgt   = (const float*)d_in[1];
  const float* wnorm = (const float*)d_in[2];
  const float* fb0_w = (const float*)d_in[3];
  const float* fb0_b = (const float*)d_in[4];
  const float* bn0_g = (const float*)d_in[5];
  const float* bn0_b = (const float*)d_in[6];
  const float* fb1_w = (const float*)d_in[7];
  const float* fb1_b = (const float*)d_in[8];
  const float* bn1_g = (const float*)d_in[9];
  const float* bn1_b = (const float*)d_in[10];
  const float* fb2_w = (const float*)d_in[11];
  const float* fb2_b = (const float*)d_in[12];
  const float* bn2_g = (const float*)d_in[13];
  const float* bn2_b = (const float*)d_in[14];
  const float* c1_w  = (const float*)d_in[15];
  const float* c1_b  = (const float*)d_in[16];
  const float* c2_w  = (const float*)d_in[17];
  const float* c2_b  = (const float*)d_in[18];
  const float* f0_w  = (const float*)d_in[19];
  const float* f0_b  = (const float*)d_in[20];
  const float* f1_w  = (const float*)d_in[21];
  const float* f1_b  = (const float*)d_in[22];
  const float* f2_w  = (const float*)d_in[23];
  const float* f2_b  = (const float*)d_in[24];
  float* ws = (float*)d_ws;
  float* out = (float*)d_out;

  k_xmom   <<<NBX, 256, 0, stream>>>(x, ws + OFF_XP);
  k_bn0fold<<<1, 64, 0, stream>>>(ws + OFF_XP, fb0_w, fb0_b, bn0_g, bn0_b,
                                  ws + OFF_W0F, ws + OFF_B0F);
  k_fb0    <<<NS / 4, 256, 0, stream>>>(x, ws + OFF_W0F, ws + OFF_B0F, ws + OFF_R0);
  k_fb1    <<<NS / 16, 256, 0, stream>>>(ws + OFF_R0, fb1_w, ws + OFF_Z1);
  k_z1p    <<<NBZ, 256, 0, stream>>>(ws + OFF_Z1, ws + OFF_ZPS, ws + OFF_ZPQ);
  k_bn1fold<<<1, 128, 0, stream>>>(ws + OFF_ZPS, ws + OFF_ZPQ, fb1_b, bn1_g, bn1_b,
                                   ws + OFF_S1, ws + OFF_B1F);
  k_z2     <<<NS / 2, 256, 0, stream>>>(ws + OFF_Z1, ws + OFF_S1, ws + OFF_B1F, fb2_w,
                                        ws + OFF_Z2, ws + OFF_Z2P);
  k_bn2fold<<<1, 256, 0, stream>>>(ws + OFF_Z2P, fb2_b, bn2_g, bn2_b, ws + OFF_SB2);
  k_ht     <<<NS / 256, 256, 0, stream>>>(ws + OFF_Z2, ws + OFF_SB2, ws + OFF_HT);
  k_mh     <<<(Bn * Vn + 255) / 256, 256, 0, stream>>>(ws + OFF_HT, ws + OFF_MH);
  k_cm     <<<(Bn * CCn * Vn * Vn) / 256, 256, 0, stream>>>(ws + OFF_MH, c1_w, c1_b,
                                                            c2_w, c2_b, ws + OFF_CM);
  k_wsum   <<<(CCn * TSn + 255) / 256, 256, 0, stream>>>(wgt, ws + OFF_WSUM);
  k_gcs    <<<(Bn * Vn * Vn + 255) / 256, 256, 0, stream>>>(ws + OFF_CM, ws + OFF_WSUM,
                                                            out, ws + OFF_N1);
  k_pan    <<<1, 256, 0, stream>>>(ws + OFF_N1, out + 40001);
  k_reg    <<<1, 256, 0, stream>>>(f0_w, f1_w, f2_w, out + 40002);
  k_fcp    <<<Bn * CCn, 256, 0, stream>>>(ws + OFF_HT, wgt, wnorm, ws + OFF_CM, ws + OFF_P1);
  k_f0     <<<Bn * TSn, 256, 0, stream>>>(ws + OFF_P1, f0_w, f0_b, ws + OFF_Q0);
  k_f1     <<<Bn * TSn, 256, 0, stream>>>(ws + OFF_Q0, f1_w, f1_b, ws + OFF_Q1);
  k_f2loss <<<Bn * TSn, 128, 0, stream>>>(ws + OFF_Q1, f2_w, f2_b, ws + OFF_HT, ws + OFF_LP);
  k_lossfin<<<1, 256, 0, stream>>>(ws + OFF_LP, out + 40000);
}